// Group_12738873000654
// MI455X (gfx1250) — compile-verified
//
#include <hip/hip_runtime.h>

// ---------------------------------------------------------------------------
// Problem constants (match reference)
// ---------------------------------------------------------------------------
#define B_ 8
#define N_ 16384
#define C_ 512          // NUM_CENTERS
#define K_ 32           // NUM_NEIGHBORS

typedef __attribute__((ext_vector_type(2))) float v2f;
typedef __attribute__((ext_vector_type(8))) float v8f;

#define INF_F (__builtin_inff())

// ---------------------------------------------------------------------------
// DS_SWIZZLE_B32 helpers: compile-time lane-permute patterns, no index VGPR.
// group-of-32 mode: offset[14:10]=xor, [9:5]=or, [4:0]=and
// ---------------------------------------------------------------------------
template<int PAT>
__device__ __forceinline__ float swzf(float v) {
    return __int_as_float(__builtin_amdgcn_ds_swizzle(__float_as_int(v), PAT));
}
template<int PAT>
__device__ __forceinline__ int swzi(int v) {
    return __builtin_amdgcn_ds_swizzle(v, PAT);
}
#define SWZ_REV  ((31 << 10) | 0x1f)   // lane <-> 31-lane
#define SWZ_BC31 (31 << 5)             // broadcast lane 31

// compare-exchange with partner lane^J; direction folded into a sign so it
// lowers to one v_cmp_lt_f32 instead of a dual-predicate select chain.
template<int J>
__device__ __forceinline__ void cx(float& v, int& i, bool keepMin) {
    constexpr int pat = (J << 10) | 0x1f;
    const float pv = swzf<pat>(v);
    const int   pi = swzi<pat>(i);
    const float s  = keepMin ? 1.0f : -1.0f;
    if (s * pv < s * v) { v = pv; i = pi; }
}

// full ascending bitonic sort of 32 (value,index) pairs, one per lane
__device__ __forceinline__ void sort32(float& v, int& i, int lane) {
    cx<1>(v, i, ((lane & 2) == 0)  == ((lane & 1) == 0));
    cx<2>(v, i, ((lane & 4) == 0)  == ((lane & 2) == 0));
    cx<1>(v, i, ((lane & 4) == 0)  == ((lane & 1) == 0));
    cx<4>(v, i, ((lane & 8) == 0)  == ((lane & 4) == 0));
    cx<2>(v, i, ((lane & 8) == 0)  == ((lane & 2) == 0));
    cx<1>(v, i, ((lane & 8) == 0)  == ((lane & 1) == 0));
    cx<8>(v, i, ((lane & 16) == 0) == ((lane & 8) == 0));
    cx<4>(v, i, ((lane & 16) == 0) == ((lane & 4) == 0));
    cx<2>(v, i, ((lane & 16) == 0) == ((lane & 2) == 0));
    cx<1>(v, i, ((lane & 16) == 0) == ((lane & 1) == 0));
    cx<16>(v, i, (lane & 16) == 0);
    cx<8> (v, i, (lane & 8)  == 0);
    cx<4> (v, i, (lane & 4)  == 0);
    cx<2> (v, i, (lane & 2)  == 0);
    cx<1> (v, i, (lane & 1)  == 0);
}

// merge sorted-ascending resident list with sorted-ascending batch,
// keep the 32 smallest, resident stays sorted ascending
__device__ __forceinline__ void mergeTopK(float& rv, int& ri, float bv, int bi, int lane) {
    const float fv = swzf<SWZ_REV>(bv);    // reversed batch
    const int   fi = swzi<SWZ_REV>(bi);
    if (fv < rv) { rv = fv; ri = fi; }     // elementwise min -> bitonic, holds global top-32
    cx<16>(rv, ri, (lane & 16) == 0);      // bitonic cleanup -> ascending
    cx<8> (rv, ri, (lane & 8)  == 0);
    cx<4> (rv, ri, (lane & 4)  == 0);
    cx<2> (rv, ri, (lane & 2)  == 0);
    cx<1> (rv, ri, (lane & 1)  == 0);
}

// ===========================================================================
// Kernel 1: farthest point sampling, one workgroup per cloud.
// LDS holds float4{x,y,z,mindist} per point (256 KB of the 320 KB WGP LDS):
// the 512-step sequential loop costs one ds_load_b128 + one 4B writeback per
// point. 1024 threads = 32 waves; wave32 shfl argmax tree + LDS reduction.
// ===========================================================================
#define FPS_THREADS 1024
#define PTS_PER_THREAD (N_ / FPS_THREADS)   // 16

extern "C" __global__ void __launch_bounds__(FPS_THREADS)
fps_kernel(const float* __restrict__ points,   // (B,N,3)
           const int*   __restrict__ lengths,  // (B,)
           float*       __restrict__ centers)  // (B,C,3)
{
    const int b      = blockIdx.x;
    const int tid    = threadIdx.x;
    const int lane   = tid & 31;
    const int wave   = tid >> 5;
    const int length = lengths[b];
    const float* pts = points + (size_t)b * N_ * 3;

    extern __shared__ float4 pt4[];              // N_ x {x,y,z,mindist}
    float* sbest = (float*)(pt4 + N_);           // 32 per-wave best value
    int*   sidx  = (int*)(sbest + 32);           // 32 per-wave best index
    int*   ssel  = sidx + 32;                    // winner index

    // Fill LDS once
    for (int i = tid; i < N_; i += FPS_THREADS) {
        float4 q;
        q.x = pts[i * 3 + 0];
        q.y = pts[i * 3 + 1];
        q.z = pts[i * 3 + 2];
        q.w = INF_F;
        pt4[i] = q;
    }
    __syncthreads();

    int last = 0;   // pytorch3d FPS starts at index 0
    for (int c = 0; c < C_; ++c) {
        // broadcast coords of most recently selected point (LDS broadcast read)
        const float4 cp = pt4[last];
        if (tid == 0) {
            const size_t o = ((size_t)b * C_ + c) * 3;
            centers[o + 0] = cp.x; centers[o + 1] = cp.y; centers[o + 2] = cp.z;
        }

        // update running mindist for my 16 points; track local argmax
        float best = -INF_F;
        int   bidx = 0;
#pragma unroll
        for (int k = 0; k < PTS_PER_THREAD; ++k) {
            const int i = tid + k * FPS_THREADS;     // 16B lane stride: conflict-free
            const float4 q = pt4[i];
            const float dx = q.x - cp.x, dy = q.y - cp.y, dz = q.z - cp.z;
            const float d  = dx * dx + dy * dy + dz * dz;
            const float m  = fminf(q.w, d);
            pt4[i].w = m;
            const float v = (i < length) ? m : -INF_F;
            if (v > best) { best = v; bidx = i; }
        }
        // wave32 argmax tree
#pragma unroll
        for (int off = 16; off > 0; off >>= 1) {
            const float ov = __shfl_down(best, off, 32);
            const int   oi = __shfl_down(bidx, off, 32);
            if (ov > best) { best = ov; bidx = oi; }
        }
        if (lane == 0) { sbest[wave] = best; sidx[wave] = bidx; }
        __syncthreads();
        if (wave == 0) {
            float v  = sbest[lane];
            int   ix = sidx[lane];
#pragma unroll
            for (int off = 16; off > 0; off >>= 1) {
                const float ov = __shfl_down(v,  off, 32);
                const int   oi = __shfl_down(ix, off, 32);
                if (ov > v) { v = ov; ix = oi; }
            }
            if (lane == 0) ssel[0] = ix;
        }
        __syncthreads();
        last = ssel[0];
    }
}

// ===========================================================================
// Kernel 2: brute-force KNN (K=32) + group gather.
// d^2 = |c|^2 + |p|^2 - 2 c.p is computed ENTIRELY inside
// V_WMMA_F32_16X16X4_F32 (A = -2*centers, C preloaded with |c|^2+|p|^2,
// +inf poison for invalid points). Selection: warp-bitonic top-32 via
// ds_swizzle, with float4 batched ballot early-out. One block = 16 centers
// = 16 waves; d^2 tile double-buffered in LDS -> one barrier per chunk.
// ===========================================================================
#define TILE_C     16
#define PCHUNK     512
#define LDP        516                   // pad: 516 % 64 = 4 -> rows 8 apart hit disjoint banks
#define KNN_WAVES  16
#define KNN_THREADS (KNN_WAVES * 32)
#define NCHUNKS    (N_ / PCHUNK)

extern "C" __global__ void __launch_bounds__(KNN_THREADS)
knn_group_kernel(const float* __restrict__ points,   // (B,N,3)
                 const int*   __restrict__ lengths,  // (B,)
                 const float* __restrict__ centers,  // (B,C,3)
                 float*       __restrict__ groups)   // (B,C,K,3)
{
    const int blk    = blockIdx.x;
    const int b      = blk / (C_ / TILE_C);
    const int c0     = (blk % (C_ / TILE_C)) * TILE_C;
    const int tid    = threadIdx.x;
    const int lane   = tid & 31;
    const int wave   = tid >> 5;
    const int length = lengths[b];
    const float* pts = points + (size_t)b * N_ * 3;

    __shared__ float d2s[2][TILE_C * LDP];           // double-buffered 16x512 d^2 tile
    __shared__ float scx[TILE_C], scy[TILE_C], scz[TILE_C], scn2[TILE_C];

    if (tid < TILE_C) {
        const size_t o = ((size_t)b * C_ + c0 + tid) * 3;
        const float x = centers[o + 0], y = centers[o + 1], z = centers[o + 2];
        scx[tid] = x; scy[tid] = y; scz[tid] = z;
        scn2[tid] = x * x + y * y + z * z;
    }
    __syncthreads();

    // A fragment (16x4 f32) scaled by -2:
    // lanes 0-15  hold (K=0,K=1)=(-2x,-2y) of row M=lane,
    // lanes 16-31 hold (K=2,K=3)=(-2z,  0) of row M=lane-16.
    const int  half = (lane < 16) ? 0 : 1;
    const int  mrow = lane & 15;
    v2f afrag;
    afrag.x = -2.0f * (half ? scz[mrow] : scx[mrow]);
    afrag.y = -2.0f * (half ? 0.0f      : scy[mrow]);

    // per-lane copy of the 8 center norms this lane accumulates (broadcast reads)
    float c2r[8];
#pragma unroll
    for (int r = 0; r < 8; ++r) c2r[r] = scn2[r + (half ? 8 : 0)];

    // LDS row offset this lane writes its 8 accumulator values into
    const int rowBase = (half ? 8 : 0) * LDP;

    // resident top-32: one (d2, idx) per lane, sorted ascending
    float rv  = INF_F;
    int   ri  = 0;
    float kth = INF_F;                               // current 32nd-smallest (lane-31 value)

    const int colw = wave * (PCHUNK / KNN_WAVES);    // this wave's 32-col slice

    for (int chunk = 0; chunk < NCHUNKS; ++chunk) {
        const int cb  = chunk * PCHUNK;
        float* buf = d2s[chunk & 1];

        // ---- phase 1: d^2 tile, 2 column-tiles of 16 per wave ----
#pragma unroll
        for (int tt = 0; tt < 2; ++tt) {
            const int col = colw + tt * 16 + mrow;
            const int g   = cb + col;
            const float px = pts[g * 3 + 0];
            const float py = pts[g * 3 + 1];
            const float pz = pts[g * 3 + 2];
            // |p|^2, poisoned to +inf for out-of-length columns
            float pbase = px * px + py * py + pz * pz;
            pbase += (g >= length) ? INF_F : 0.0f;
            // B fragment (4x16 f32), mirror of A layout
            v2f bfrag;
            bfrag.x = half ? pz   : px;
            bfrag.y = half ? 0.0f : py;
            // C preloaded with |c|^2 + |p|^2 (+inf)  ->  D is the final d^2
            v8f acc;
#pragma unroll
            for (int r = 0; r < 8; ++r) acc[r] = c2r[r] + pbase;
            acc = __builtin_amdgcn_wmma_f32_16x16x4_f32(
                false, afrag, false, bfrag, (short)0, acc, false, false);
#pragma unroll
            for (int r = 0; r < 8; ++r)
                buf[rowBase + r * LDP + col] = acc[r];
        }
        // prefetch next chunk's points while phase 2 runs (gfx1250 global_prefetch)
        if (chunk + 1 < NCHUNKS) {
            const int gn = cb + PCHUNK + colw + mrow;
            __builtin_prefetch(&pts[gn * 3], 0, 3);
        }
        __syncthreads();   // producers done with buf; double-buffer covers reuse

        // ---- phase 2: wave w streams row w into its bitonic top-32 ----
        // lane holds 4 consecutive d^2 values per ds_load_b128 (16B stride:
        // conflict-free); one combined ballot pre-test per 128 candidates.
        const float* row = buf + wave * LDP;
#pragma unroll 1
        for (int base = 0; base < PCHUNK; base += 128) {
            const float4 q4 = *(const float4*)(row + base + lane * 4);
            const float q[4] = { q4.x, q4.y, q4.z, q4.w };
            const bool any4 = (q[0] < kth) | (q[1] < kth) | (q[2] < kth) | (q[3] < kth);
            if (__ballot(any4)) {
#pragma unroll
                for (int cc = 0; cc < 4; ++cc) {
                    const float bv = q[cc];
                    if (__ballot(bv < kth)) {
                        float sv = bv;
                        int   bi = cb + base + lane * 4 + cc;
                        sort32(sv, bi, lane);
                        mergeTopK(rv, ri, sv, bi, lane);
                        kth = swzf<SWZ_BC31>(rv);
                    }
                }
            }
        }
    }

    // ---- output: lane = neighbor rank (ascending d2, matches top_k(-d2)) ----
    const int   c  = c0 + wave;
    const float cx0 = scx[wave], cy0 = scy[wave], cz0 = scz[wave];
    const int   idx = ri;
    const float gx = pts[idx * 3 + 0] - cx0;
    const float gy = pts[idx * 3 + 1] - cy0;
    const float gz = pts[idx * 3 + 2] - cz0;
    const size_t o = (((size_t)b * C_ + c) * K_ + lane) * 3;
    groups[o + 0] = gx; groups[o + 1] = gy; groups[o + 2] = gz;
}

// ===========================================================================
// Host launcher
// ===========================================================================
extern "C" void kernel_launch(void* const* d_in, const int* in_sizes, int n_in,
                              void* d_out, int out_size, void* d_ws, size_t ws_size,
                              hipStream_t stream)
{
    (void)in_sizes; (void)n_in; (void)out_size; (void)d_ws; (void)ws_size;

    const float* points  = (const float*)d_in[0];   // (8,16384,3) f32
    const int*   lengths = (const int*)  d_in[1];   // (8,) i32

    float* groupsOut  = (float*)d_out;                                  // B*C*K*3
    float* centersOut = (float*)d_out + (size_t)B_ * C_ * K_ * 3;       // B*C*3

    // FPS: one workgroup per cloud; 256 KB + change of dynamic LDS
    const size_t fpsLds = (size_t)N_ * sizeof(float4) + 80 * sizeof(float);
    fps_kernel<<<dim3(B_), dim3(FPS_THREADS), fpsLds, stream>>>(
        points, lengths, centersOut);

    // KNN + gather: 16 centers per block
    const int knnBlocks = B_ * (C_ / TILE_C);   // 256
    knn_group_kernel<<<dim3(knnBlocks), dim3(KNN_THREADS), 0, stream>>>(
        points, lengths, centersOut, groupsOut);
}